// LightGCN_W_30313879175390
// MI455X (gfx1250) — compile-verified
//
#include <hip/hip_runtime.h>

// ---------------------------------------------------------------------------
// LightGCN window encode for MI455X (gfx1250, wave32).
// Memory-bound scatter SpMM: all randomly-accessed state (x ping-pong 76.8MB +
// norm 16MB) fits in the 192MB L2, so gathers + global_atomic_add_f32 run out
// of L2; only the 32MB edge stream hits HBM each layer. Final *0.25 scaling is
// done exactly on the matrix pipe via chained V_WMMA_F32_16X16X4_F32.
// ---------------------------------------------------------------------------

#define N_USERS   100000
#define N_ITEMS   50000
#define N_NODES   150000          // N_USERS + N_ITEMS
#define DIM       64
#define N_EDGES   4000000
#define NUM_LAYERS 3

#define ND4       2400000         // N_NODES*DIM/4
#define UD4       1600000         // N_USERS*DIM/4
#define DEG_PAD   150016          // N_NODES padded to /16

typedef __attribute__((ext_vector_type(2))) float v2f;
typedef __attribute__((ext_vector_type(8))) float v8f;

// ---------------- elementwise helpers (float4, grid covers exactly) --------
__global__ void zero_f4(float4* __restrict__ p, int n4) {
  int i = blockIdx.x * blockDim.x + threadIdx.x;
  if (i < n4) p[i] = make_float4(0.f, 0.f, 0.f, 0.f);
}

__global__ void acc_f4(float4* __restrict__ out, const float4* __restrict__ x, int n4) {
  int i = blockIdx.x * blockDim.x + threadIdx.x;
  if (i < n4) {
    float4 o = out[i], v = x[i];
    o.x += v.x; o.y += v.y; o.z += v.z; o.w += v.w;
    out[i] = o;
  }
}

// x_cur = concat(user, item); out(acc) starts as x0
__global__ void init_x(const float4* __restrict__ user, const float4* __restrict__ item,
                       float4* __restrict__ x, float4* __restrict__ out) {
  int i = blockIdx.x * blockDim.x + threadIdx.x;
  if (i < ND4) {
    float4 v = (i < UD4) ? user[i] : item[i - UD4];
    x[i] = v;
    out[i] = v;
  }
}

// ---------------- degree / normalization ----------------------------------
__global__ void deg_count(const int* __restrict__ col, float* __restrict__ deg) {
  int e = blockIdx.x * blockDim.x + threadIdx.x;
  if (e < N_EDGES) unsafeAtomicAdd(&deg[col[e]], 1.0f);   // global_atomic_add_f32
}

__global__ void deg_to_dinv(float* __restrict__ deg) {
  int i = blockIdx.x * blockDim.x + threadIdx.x;
  if (i < N_NODES) {
    float d = deg[i];
    deg[i] = (d > 0.f) ? rsqrtf(d) : 0.f;   // matches where(deg>0, rsqrt(deg), 0)
  }
}

__global__ void edge_norm(const int* __restrict__ row, const int* __restrict__ col,
                          const float* __restrict__ dinv, float* __restrict__ norm) {
  int e = blockIdx.x * blockDim.x + threadIdx.x;
  if (e < N_EDGES) norm[e] = dinv[row[e]] * dinv[col[e]];
}

// ---------------- the hot loop: per-edge scatter ---------------------------
// 16 lanes per edge, one float4 per lane -> one coalesced 256B gather of
// x_in[row], then 64 global_atomic_add_f32 into x_out[col] (L2-resident).
__global__ void spmm_scatter(const int* __restrict__ row, const int* __restrict__ col,
                             const float* __restrict__ norm,
                             const float* __restrict__ xin, float* __restrict__ xout) {
  int t = blockIdx.x * blockDim.x + threadIdx.x;
  int e = t >> 4;
  int l = t & 15;
  if (e >= N_EDGES) return;
  int r = row[e];
  int c = col[e];
  float w = norm[e];
  const float4* xr = (const float4*)(xin + (size_t)r * DIM);
  float4 v = xr[l];
  float* dst = xout + (size_t)c * DIM + l * 4;
  unsafeAtomicAdd(dst + 0, w * v.x);
  unsafeAtomicAdd(dst + 1, w * v.y);
  unsafeAtomicAdd(dst + 2, w * v.z);
  unsafeAtomicAdd(dst + 3, w * v.w);
}

// ---------------- finalize: out *= 0.25 on the WMMA pipe (exact in f32) ----
// One wave per 256-float chunk viewed as a 16x16 row-major tile X.
// D = sum_k A_k(16x4) * B_k(4x16) with A_k = X[:,4k:4k+4] and B_k the k-th
// band of 0.25*I16.  Multiplication by 0.25 and additions with zero are exact.
// Layouts (wave32): A 16x4 -> lane m holds K = j + 2*(lane>=16) in vgpr j.
//                   B 4x16 -> lane n holds K = j + 2*(lane>=16) in vgpr j.
//                   C/D    -> lane n, vgpr v holds M = v + 8*(lane>=16).
__global__ void finalize_scale_wmma(float* __restrict__ out) {
  int wave = (int)((blockIdx.x * blockDim.x + threadIdx.x) >> 5);
  if (wave >= ND4 / 64) return;            // wave-uniform; EXEC stays all-ones
  int lane = threadIdx.x & 31;
  int half = lane >> 4;                    // 0: lanes 0-15, 1: lanes 16-31
  int l16  = lane & 15;
  float* chunk = out + (size_t)wave * 256;

  v8f c = {0.f, 0.f, 0.f, 0.f, 0.f, 0.f, 0.f, 0.f};
#pragma unroll
  for (int k = 0; k < 4; ++k) {
    const float* ap = chunk + l16 * 16 + 4 * k + 2 * half;
    v2f a;
    a.x = ap[0];
    a.y = ap[1];
    v2f b;
    b.x = (l16 == 4 * k + 2 * half + 0) ? 0.25f : 0.0f;
    b.y = (l16 == 4 * k + 2 * half + 1) ? 0.25f : 0.0f;
    c = __builtin_amdgcn_wmma_f32_16x16x4_f32(
        /*neg_a=*/false, a, /*neg_b=*/false, b,
        /*c_mod=*/(short)0, c, /*reuse_a=*/false, /*reuse_b=*/false);
  }
#pragma unroll
  for (int v = 0; v < 8; ++v)
    chunk[(v + 8 * half) * 16 + l16] = c[v];
}

// ---------------------------------------------------------------------------
extern "C" void kernel_launch(void* const* d_in, const int* in_sizes, int n_in,
                              void* d_out, int out_size, void* d_ws, size_t ws_size,
                              hipStream_t stream) {
  const float* user = (const float*)d_in[0];
  const float* item = (const float*)d_in[1];
  const int*   row  = (const int*)d_in[2];          // edge_index[0]
  const int*   col  = row + N_EDGES;                // edge_index[1]
  float* out = (float*)d_out;                       // [N_NODES, 64]

  // workspace layout (floats): deg/dinv | norm | x ping | x pong  (~93.4 MB)
  float* ws   = (float*)d_ws;
  float* deg  = ws;                                 // DEG_PAD
  float* norm = ws + DEG_PAD;                       // N_EDGES
  float* xa   = norm + N_EDGES;                     // N_NODES*DIM
  float* xb   = xa + (size_t)N_NODES * DIM;         // N_NODES*DIM

  // 1) degrees (counted at col) -> dinv (in place) -> per-edge norm
  zero_f4<<<(DEG_PAD / 4 + 255) / 256, 256, 0, stream>>>((float4*)deg, DEG_PAD / 4);
  deg_count<<<(N_EDGES + 255) / 256, 256, 0, stream>>>(col, deg);
  deg_to_dinv<<<(N_NODES + 255) / 256, 256, 0, stream>>>(deg);
  edge_norm<<<(N_EDGES + 255) / 256, 256, 0, stream>>>(row, col, deg, norm);

  // 2) x0 = concat(user, item); acc(out) = x0
  init_x<<<(ND4 + 255) / 256, 256, 0, stream>>>(
      (const float4*)user, (const float4*)item, (float4*)xa, (float4*)out);

  // 3) three propagation rounds, accumulating into out
  float* xc = xa;
  float* xn = xb;
  for (int l = 0; l < NUM_LAYERS; ++l) {
    zero_f4<<<ND4 / 256, 256, 0, stream>>>((float4*)xn, ND4);
    spmm_scatter<<<(N_EDGES * 16) / 256, 256, 0, stream>>>(row, col, norm, xc, xn);
    acc_f4<<<ND4 / 256, 256, 0, stream>>>((float4*)out, (const float4*)xn, ND4);
    float* t = xc; xc = xn; xn = t;
  }

  // 4) out *= 0.25 via chained f32 WMMAs (37500 waves, 4 waves/block)
  finalize_scale_wmma<<<(ND4 / 64 + 3) / 4, 128, 0, stream>>>(out);
}